// GraphSAGE_DGL_55594056680296
// MI455X (gfx1250) — compile-verified
//
#include <hip/hip_runtime.h>

typedef float v2f __attribute__((ext_vector_type(2)));
typedef float v8f __attribute__((ext_vector_type(8)));

// ---------------------------------------------------------------------------
// Grid-stride zero fill (graph-capture-safe replacement for hipMemsetAsync)
// ---------------------------------------------------------------------------
__global__ void zero_f32(float* __restrict__ p, long n) {
  long i = (long)blockIdx.x * blockDim.x + threadIdx.x;
  long stride = (long)gridDim.x * blockDim.x;
  for (; i < n; i += stride) p[i] = 0.0f;
}

// ---------------------------------------------------------------------------
// In-degree as float (layer invariant; computed once)
// ---------------------------------------------------------------------------
__global__ void degree_kernel(const int* __restrict__ dst,
                              float* __restrict__ deg, int E) {
  int e = blockIdx.x * blockDim.x + threadIdx.x;
  if (e < E) atomicAdd(&deg[dst[e]], 1.0f);
}

// ---------------------------------------------------------------------------
// Edge scatter-add: one wave32 per edge, 128-float rows.
// Row read is a fully-coalesced 512B burst (global_load_b128 per lane);
// accumulation via global_atomic_add_f32 (stays in L2 on MI455X: 192MB L2
// holds the whole 25.6MB destination).
// ---------------------------------------------------------------------------
__global__ void scatter_add128(const float* __restrict__ h,
                               const int* __restrict__ src,
                               const int* __restrict__ dst,
                               float* __restrict__ agg, int E) {
  int tid = blockIdx.x * blockDim.x + threadIdx.x;
  int e = tid >> 5;
  int lane = tid & 31;
  if (e >= E) return;
  int s = src[e];
  int t = dst[e];
  const float4 v =
      *reinterpret_cast<const float4*>(h + (size_t)s * 128 + lane * 4);
  float* a = agg + (size_t)t * 128 + lane * 4;
  atomicAdd(a + 0, v.x);
  atomicAdd(a + 1, v.y);
  atomicAdd(a + 2, v.z);
  atomicAdd(a + 3, v.w);
}

// ---------------------------------------------------------------------------
// Fused SAGE layer GEMM using V_WMMA_F32_16X16X4_F32 (wave32):
//   out = act( h @ Wself + b + (agg * 1/max(deg,1)) @ Wneigh )
//
// Block = 256 threads = 8 waves. Each block owns a 16-wide output-column
// slab; Wself/Wneigh column blocks (K x 16 each, K<=128) are staged in LDS
// and shared across the 8 waves. Each wave computes one 16x16 C tile.
//
// Fragment layouts per CDNA5 ISA 7.12.2:
//   A (16x4 f32, 2 VGPRs): lane L -> row M = L&15, K pair = (L>>4)*2
//   B (4x16 f32, 2 VGPRs): lane L -> col N = L&15, K pair = (L>>4)*2
//   C/D (16x16 f32, 8 VGPRs): vgpr j, lane L -> M = j + 8*(L>=16), N = L&15
// EXEC is kept all-1s around the WMMA (wave-uniform guards only; N%16==0).
// ---------------------------------------------------------------------------
template <bool RELU>
__global__ __launch_bounds__(256) void sage_gemm_wmma(
    const float* __restrict__ h, const float* __restrict__ agg,
    const float* __restrict__ deg, const float* __restrict__ wself,
    const float* __restrict__ wneigh, const float* __restrict__ bias,
    float* __restrict__ out, int nNodes, int K, int NCOL) {
  __shared__ float lws[128 * 16];
  __shared__ float lwn[128 * 16];

  const int colBase = blockIdx.y * 16;

  // Stage weight column blocks into LDS (all 256 threads cooperate).
  for (int idx = threadIdx.x; idx < K * 16; idx += 256) {
    int kk = idx >> 4;
    int cc = idx & 15;
    lws[idx] = wself[(size_t)kk * NCOL + colBase + cc];
    lwn[idx] = wneigh[(size_t)kk * NCOL + colBase + cc];
  }
  __syncthreads();

  const int wave = threadIdx.x >> 5;
  const int lane = threadIdx.x & 31;
  const int tileRow = (blockIdx.x * 8 + wave) * 16;
  if (tileRow >= nNodes) return;  // wave-uniform: EXEC stays all-1s

  const int col = lane & 15;            // A row offset == B/C col offset
  const int pairBase = (lane >> 4) * 2; // 0 for lanes 0-15, 2 for 16-31
  const int r = tileRow + col;          // A-fragment row for this lane

  const float rd = 1.0f / fmaxf(deg[r], 1.0f);  // fold mean into A load
  const float* hrow = h + (size_t)r * K;
  const float* arow = agg + (size_t)r * K;

  v8f acc = {};
  for (int k0 = 0; k0 < K; k0 += 4) {
    const int ka = k0 + pairBase;
    // A fragments: aligned 8B loads (ka is even)
    v2f as = *reinterpret_cast<const v2f*>(hrow + ka);
    v2f an = *reinterpret_cast<const v2f*>(arow + ka);
    an = an * rd;
    // B fragments from LDS
    v2f bs, bn;
    bs.x = lws[ka * 16 + col];
    bs.y = lws[(ka + 1) * 16 + col];
    bn.x = lwn[ka * 16 + col];
    bn.y = lwn[(ka + 1) * 16 + col];
    acc = __builtin_amdgcn_wmma_f32_16x16x4_f32(false, as, false, bs,
                                                (short)0, acc, false, false);
    acc = __builtin_amdgcn_wmma_f32_16x16x4_f32(false, an, false, bn,
                                                (short)0, acc, false, false);
  }

  const float bc = bias[colBase + col];
  const int rowHi = (lane >> 4) * 8;
  for (int j = 0; j < 8; ++j) {
    float v = acc[j] + bc;
    if (RELU) v = fmaxf(v, 0.0f);
    out[(size_t)(tileRow + rowHi + j) * NCOL + colBase + col] = v;
  }
}

// ---------------------------------------------------------------------------
// Host launch
// ---------------------------------------------------------------------------
extern "C" void kernel_launch(void* const* d_in, const int* in_sizes, int n_in,
                              void* d_out, int out_size, void* d_ws,
                              size_t ws_size, hipStream_t stream) {
  const float* feat = (const float*)d_in[0];
  const int* src = (const int*)d_in[1];
  const int* dst = (const int*)d_in[2];
  const float* ws0 = (const float*)d_in[3];
  const float* wn0 = (const float*)d_in[4];
  const float* b0 = (const float*)d_in[5];
  const float* ws1 = (const float*)d_in[6];
  const float* wn1 = (const float*)d_in[7];
  const float* b1 = (const float*)d_in[8];
  const float* ws2 = (const float*)d_in[9];
  const float* wn2 = (const float*)d_in[10];
  const float* b2 = (const float*)d_in[11];

  const int E = in_sizes[1];
  const int H = in_sizes[5];            // |b0|
  const int IN = in_sizes[3] / H;       // w_self0 is IN x H
  const int OUT = in_sizes[11];         // |b2|
  const int N = in_sizes[0] / IN;       // feat is N x IN

  // Workspace carve-out (256B-aligned regions)
  auto alignup = [](size_t x) { return (x + 255) & ~(size_t)255; };
  char* base = (char*)d_ws;
  size_t off = 0;
  float* deg = (float*)(base + off);
  off += alignup((size_t)N * sizeof(float));
  float* agg = (float*)(base + off);
  off += alignup((size_t)N * 128 * sizeof(float));
  float* h0 = (float*)(base + off);
  off += alignup((size_t)N * 128 * sizeof(float));
  float* h1 = (float*)(base + off);

  const dim3 blk(256);
  const int zeroGrid = 2048;
  const int edgeGrid = (E + 255) / 256;
  const int scatGrid = (int)(((size_t)E * 32 + 255) / 256);

  // Degree (once)
  zero_f32<<<zeroGrid, blk, 0, stream>>>(deg, (long)N);
  degree_kernel<<<edgeGrid, blk, 0, stream>>>(dst, deg, E);

  // ---- Layer 0: feat -> h0 (relu) ----
  zero_f32<<<zeroGrid, blk, 0, stream>>>(agg, (long)N * IN);
  scatter_add128<<<scatGrid, blk, 0, stream>>>(feat, src, dst, agg, E);
  {
    dim3 grid((N + 127) / 128, H / 16);
    sage_gemm_wmma<true><<<grid, blk, 0, stream>>>(feat, agg, deg, ws0, wn0,
                                                   b0, h0, N, IN, H);
  }

  // ---- Layer 1: h0 -> h1 (relu) ----
  zero_f32<<<zeroGrid, blk, 0, stream>>>(agg, (long)N * H);
  scatter_add128<<<scatGrid, blk, 0, stream>>>(h0, src, dst, agg, E);
  {
    dim3 grid((N + 127) / 128, H / 16);
    sage_gemm_wmma<true><<<grid, blk, 0, stream>>>(h0, agg, deg, ws1, wn1, b1,
                                                   h1, N, H, H);
  }

  // ---- Layer 2: h1 -> out (no act) ----
  zero_f32<<<zeroGrid, blk, 0, stream>>>(agg, (long)N * H);
  scatter_add128<<<scatGrid, blk, 0, stream>>>(h1, src, dst, agg, E);
  {
    dim3 grid((N + 127) / 128, OUT / 16);
    sage_gemm_wmma<false><<<grid, blk, 0, stream>>>(h1, agg, deg, ws2, wn2, b2,
                                                    (float*)d_out, N, H, OUT);
  }
}